// QKVCreator_88871463288884
// MI455X (gfx1250) — compile-verified
//
#include <hip/hip_runtime.h>
#include <hip/hip_bf16.h>
#include <math.h>
#include <stdint.h>

typedef __attribute__((ext_vector_type(2))) float v2f;
typedef __attribute__((ext_vector_type(8))) float v8f;

namespace {
constexpr int BM = 128, BN = 128, BK = 32;
constexpr int LDK = 36;              // row stride (floats): 144B rows -> 16B aligned, conflict-free frags
constexpr int TILE = BM * LDK;       // 4608 floats per tile buffer
constexpr int LDQ = 130;             // Q epilogue tile stride
constexpr int HID = 4096, SEQ = 2048, NH = 32, HD = 128;
constexpr int THREADS = 256;
}

// Async global->LDS copy of 16 bytes/lane. IOFFSET is added to BOTH the LDS
// destination and the global source address (CDNA5 ISA).
#define ASYNC_CP16(ldsOff, gOff, base, IMM)                                   \
    asm volatile("global_load_async_to_lds_b128 %0, %1, %2 offset:" #IMM      \
                 :: "v"(ldsOff), "v"(gOff), "s"(base) : "memory")

__global__ __launch_bounds__(THREADS, 1)
void qkv_rope_wmma(const float* __restrict__ x,
                   const float* __restrict__ wq, const float* __restrict__ bq,
                   const float* __restrict__ wk, const float* __restrict__ bk,
                   const float* __restrict__ wv, const float* __restrict__ bv,
                   float* __restrict__ out)
{
    extern __shared__ float smem[];
    // layout: X buf0 | X buf1 | W buf0 | W buf1   (4*TILE floats = 72 KB)

    const int z = blockIdx.z;
    const float* __restrict__ w    = (z == 0) ? wq : (z == 1) ? wk : wv;
    const float* __restrict__ bias = (z == 0) ? bq : (z == 1) ? bk : bv;

    const int head  = blockIdx.y;        // BN == head_dim -> one head per N tile
    const int nBase = head * BN;
    const int mBase = blockIdx.x * BM;   // 128 | 2048 -> single batch per tile
    const int bIdx  = mBase / SEQ;
    const int sBase = mBase % SEQ;

    const int tid  = threadIdx.x;
    const int lane = tid & 31;
    const int wid  = tid >> 5;
    const int waveM = (wid & 3) * 32;    // 4 waves across M
    const int waveN = (wid >> 2) * 64;   // 2 waves across N
    const int laneLo  = lane & 15;
    const int laneHiK = (lane >> 4) * 2; // K offset 0 or 2 for A/B fragments
    const int laneHiM = (lane >> 4) * 8; // row offset 0 or 8 for C/D tiles

    v8f acc[2][4];
    #pragma unroll
    for (int mi = 0; mi < 2; ++mi)
        #pragma unroll
        for (int ni = 0; ni < 4; ++ni)
            #pragma unroll
            for (int r = 0; r < 8; ++r)
                acc[mi][ni][r] = 0.0f;

    // Copy assignment: 2 threads per tile row, 16 floats (4 x b128) each.
    const int lrow = tid >> 1;            // 0..127
    const int lcol = (tid & 1) * 16;      // 0 or 16
    // LDS byte offsets (low 32 bits of the flat address == LDS offset)
    const uint32_t ldsX0 = (uint32_t)(uintptr_t)(smem + lrow * LDK + lcol);
    const uint32_t ldsW0 = (uint32_t)(uintptr_t)(smem + 2 * TILE + lrow * LDK + lcol);
    // 32-bit global byte offsets from the SGPR base pointers (max ~134 MB, fits)
    const uint32_t xvo = (uint32_t)((mBase + lrow) * HID + lcol) * 4u;
    const uint32_t wvo = (uint32_t)((nBase + lrow) * HID + lcol) * 4u;

    auto issue_chunk = [&](int parity, int k0) {
        const uint32_t xo = ldsX0 + (uint32_t)parity * (TILE * 4u);
        const uint32_t wo = ldsW0 + (uint32_t)parity * (TILE * 4u);
        const uint32_t gx = xvo + (uint32_t)k0 * 4u;
        const uint32_t gw = wvo + (uint32_t)k0 * 4u;
        ASYNC_CP16(xo, gx, x, 0);
        ASYNC_CP16(xo, gx, x, 16);
        ASYNC_CP16(xo, gx, x, 32);
        ASYNC_CP16(xo, gx, x, 48);
        ASYNC_CP16(wo, gw, w, 0);
        ASYNC_CP16(wo, gw, w, 16);
        ASYNC_CP16(wo, gw, w, 32);
        ASYNC_CP16(wo, gw, w, 48);
    };

    issue_chunk(0, 0);   // prologue: chunk 0 -> buffer 0

    #pragma unroll 1
    for (int k0 = 0; k0 < HID; k0 += BK) {
        const int par = (k0 / BK) & 1;
        if (k0 + BK < HID) {
            issue_chunk(par ^ 1, k0 + BK);      // background copy of next chunk
            // in-order completion: <=8 outstanding ==> current chunk landed
            asm volatile("s_wait_asynccnt 0x8" ::: "memory");
        } else {
            asm volatile("s_wait_asynccnt 0x0" ::: "memory");
        }
        __syncthreads();                        // all waves' copies visible

        const float* ldsX = smem + par * TILE;
        const float* ldsW = smem + 2 * TILE + par * TILE;

        #pragma unroll
        for (int kk = 0; kk < BK; kk += 4) {
            const int kb = kk + laneHiK;
            v2f af[2], bf[4];
            #pragma unroll
            for (int mi = 0; mi < 2; ++mi)
                af[mi] = *(const v2f*)(ldsX + (waveM + mi * 16 + laneLo) * LDK + kb);
            #pragma unroll
            for (int ni = 0; ni < 4; ++ni)
                bf[ni] = *(const v2f*)(ldsW + (waveN + ni * 16 + laneLo) * LDK + kb);
            #pragma unroll
            for (int mi = 0; mi < 2; ++mi)
                #pragma unroll
                for (int ni = 0; ni < 4; ++ni)
                    acc[mi][ni] = __builtin_amdgcn_wmma_f32_16x16x4_f32(
                        false, af[mi], false, bf[ni],
                        (short)0, acc[mi][ni], false, false);
        }
        __syncthreads();   // everyone done reading this buffer before it is refilled
    }

    // Per-lane bias (column-dependent only)
    float bcol[4];
    #pragma unroll
    for (int ni = 0; ni < 4; ++ni)
        bcol[ni] = bias[nBase + waveN + ni * 16 + laneLo];

    const int Btot = (gridDim.x * BM) / SEQ;                    // batch count
    const size_t projStride = (size_t)Btot * NH * SEQ * HD;     // floats per projection
    float* outP = out + (size_t)z * projStride
                      + ((size_t)(bIdx * NH + head)) * ((size_t)SEQ * HD);

    if (z != 0) {
        // K / V: transposed store with bias
        #pragma unroll
        for (int mi = 0; mi < 2; ++mi)
            #pragma unroll
            for (int ni = 0; ni < 4; ++ni) {
                const int d = waveN + ni * 16 + laneLo;
                #pragma unroll
                for (int r = 0; r < 8; ++r) {
                    const int s = sBase + waveM + mi * 16 + laneHiM + r;
                    outP[(size_t)s * HD + d] = acc[mi][ni][r] + bcol[ni];
                }
            }
    } else {
        // Q: bounce tile through LDS so each thread sees the (d, d+64) pair
        __syncthreads();
        float* ldsQ = smem;   // BM x LDQ floats (65 KB) fits in the 72 KB alloc
        #pragma unroll
        for (int mi = 0; mi < 2; ++mi)
            #pragma unroll
            for (int ni = 0; ni < 4; ++ni) {
                const int col = waveN + ni * 16 + laneLo;
                #pragma unroll
                for (int r = 0; r < 8; ++r) {
                    const int row = waveM + mi * 16 + laneHiM + r;
                    ldsQ[row * LDQ + col] = acc[mi][ni][r] + bcol[ni];
                }
            }
        __syncthreads();
        #pragma unroll 1
        for (int i = 0; i < (BM * HD / 2) / THREADS; ++i) {  // 32 pairs/thread
            const int p   = i * THREADS + tid;
            const int row = p >> 6;
            const int dH  = p & 63;
            const float q1 = ldsQ[row * LDQ + dH];
            const float q2 = ldsQ[row * LDQ + dH + 64];
            const int s = sBase + row;
            const float invf = powf(10000.0f, -(float)dH * (1.0f / 64.0f));
            const float ang  = (float)s * invf;
            float sn, cs;
            sincosf(ang, &sn, &cs);
            float* o = outP + (size_t)s * HD;
            o[dH]      = q1 * cs - q2 * sn;
            o[dH + 64] = q2 * cs + q1 * sn;
        }
    }
}

extern "C" void kernel_launch(void* const* d_in, const int* in_sizes, int n_in,
                              void* d_out, int out_size, void* d_ws, size_t ws_size,
                              hipStream_t stream) {
    const float* x  = (const float*)d_in[0];
    const float* wq = (const float*)d_in[1];
    const float* bq = (const float*)d_in[2];
    const float* wk = (const float*)d_in[3];
    const float* bk = (const float*)d_in[4];
    const float* wv = (const float*)d_in[5];
    const float* bv = (const float*)d_in[6];
    float* out = (float*)d_out;

    const int M = in_sizes[0] / HID;          // B*S = 8192
    dim3 grid(M / BM, HID / BN, 3);           // (64, 32, 3)
    size_t mainLds = (size_t)4 * TILE;        // double-buffered X and W
    size_t qLds    = (size_t)BM * LDQ;        // Q epilogue tile
    size_t ldsBytes = 4 * (mainLds > qLds ? mainLds : qLds);

    hipLaunchKernelGGL(qkv_rope_wmma, grid, dim3(THREADS), ldsBytes, stream,
                       x, wq, bq, wk, bk, wv, bv, out);
}